// WideNetIndexConvIndexPool_37984690766335
// MI455X (gfx1250) — compile-verified
//
#include <hip/hip_runtime.h>
#include <cstddef>

typedef __attribute__((ext_vector_type(2))) float v2f;
typedef __attribute__((ext_vector_type(8))) float v8f;

#define BATCH 256

// ---------------------------------------------------------------------------
// WMMA wrapper: D = A(16x4) * B(4x16) + C, fp32, wave32
// ---------------------------------------------------------------------------
__device__ __forceinline__ v8f wmma4(v2f a, v2f b, v8f c) {
  return __builtin_amdgcn_wmma_f32_16x16x4_f32(
      /*neg_a=*/false, a, /*neg_b=*/false, b,
      /*c_mod=*/(short)0, c, /*reuse_a=*/false, /*reuse_b=*/false);
}

// ---------------------------------------------------------------------------
// Transpose x[B, C, N] (fp32) -> xT[C, N, B]
// ---------------------------------------------------------------------------
__global__ __launch_bounds__(256)
void k_transpose(const float* __restrict__ x, float* __restrict__ xT,
                 int C, int N) {
  const int cn = blockIdx.x;          // c*N + n
  const int b  = threadIdx.x;
  const int c  = cn / N;
  const int n  = cn - c * N;
  xT[(size_t)cn * BATCH + b] = x[((size_t)b * C + c) * N + n];
}

// ---------------------------------------------------------------------------
// BatchNorm statistics: per channel over N*B elements of a[C, N, B].
// Produces fused scale/shift:  y = scale[c]*x + shift[c]
// One block (256 threads) per channel.
// ---------------------------------------------------------------------------
__global__ __launch_bounds__(256)
void k_bn_stats(const float* __restrict__ a, int NB,
                const float* __restrict__ g, const float* __restrict__ bt,
                float* __restrict__ scale, float* __restrict__ shift) {
  __shared__ float ssum[256];
  __shared__ float ssq[256];
  const int c = blockIdx.x;
  const float* p = a + (size_t)c * NB;
  float s = 0.f, q = 0.f;
  for (int i = threadIdx.x; i < NB; i += 256) {
    float v = p[i];
    s += v;
    q = fmaf(v, v, q);
  }
  ssum[threadIdx.x] = s;
  ssq[threadIdx.x]  = q;
  __syncthreads();
  for (int st = 128; st > 0; st >>= 1) {
    if (threadIdx.x < st) {
      ssum[threadIdx.x] += ssum[threadIdx.x + st];
      ssq[threadIdx.x]  += ssq[threadIdx.x + st];
    }
    __syncthreads();
  }
  if (threadIdx.x == 0) {
    const float invN = 1.0f / (float)NB;
    const float m    = ssum[0] * invN;
    const float var  = ssq[0] * invN - m * m;
    const float inv  = rsqrtf(var + 1e-5f);
    const float gc   = g[c];
    scale[c] = gc * inv;
    shift[c] = bt[c] - gc * m * inv;
  }
}

// ---------------------------------------------------------------------------
// IndexedConv via WMMA.
//   src : [C_in, N_in, BATCH]  (BN+ReLU fused on read when MODE==1)
//   idx : [N_out, KW]
//   w   : [C_out, C_in, KW]
//   dst : [C_out, N_out, BATCH] = conv + bias (+ resid)
//
// Reduction is k-outer / channel-inner: neighbor index uniform per step,
// per-lane channel is c = cb + 2*hi + {0,1}.  The channel loop is SPLIT:
//   * main loop over full 4-channel blocks: no validity masks, no clamps,
//     affine addressing (pure pointer bumps a0 += 4*crow, woff += 4*KW);
//     only the uniform m<0 cndmask pair remains per step
//   * one tail block handles C_in % 4 with clamp+mask (runs once)
// Weights are never masked (activation zero-mask makes A*0 == 0; garbage
// output rows are never stored).  One wave owns one (node, 16-batch tile)
// and all OT output-channel tiles, sharing gathered-activation loads.
// ---------------------------------------------------------------------------
template <int KW, int MODE, int OT>
__global__ __launch_bounds__(128)
void k_ic_conv(const float* __restrict__ src, const int* __restrict__ idx,
               const float* __restrict__ w, const float* __restrict__ bias,
               const float* __restrict__ bscale, const float* __restrict__ bshift,
               const float* __restrict__ resid, float* __restrict__ dst,
               int C_in, int N_in, int C_out, int N_out) {
  const int n    = blockIdx.x;
  const int b0   = (blockIdx.y * 4 + threadIdx.y) * 16;
  const int lane = threadIdx.x;
  const int lo   = lane & 15;
  const int hi   = lane >> 4;
  const int bN   = b0 + lo;

  // Per-neighbor activation base pointers (node index is uniform).
  const float* sp[KW];
  bool mk[KW];
#pragma unroll
  for (int k = 0; k < KW; ++k) {
    const int m = idx[n * KW + k];
    mk[k] = (m >= 0);
    sp[k] = src + (size_t)(mk[k] ? m : 0) * BATCH + bN;
  }

  // Per-tile weight row base (clamped; garbage rows are never stored).
  const float* wt[OT];
#pragma unroll
  for (int t = 0; t < OT; ++t) {
    const int oA = t * 16 + lo;
    wt[t] = w + (size_t)(oA < C_out ? oA : C_out - 1) * C_in * KW;
  }

  v8f acc[OT] = {};
  const size_t crow   = (size_t)N_in * BATCH;   // stride between channels
  const int    cbFull = C_in & ~3;

  // ---- main loop: full 4-channel blocks, mask/clamp free, affine ----------
  size_t a0    = (size_t)(hi << 1) * crow;
  size_t a1    = a0 + crow;
  int    woff0 = (hi << 1) * KW;
  int    woff1 = woff0 + KW;

  for (int cb = 0; cb < cbFull; cb += 4) {
    float sc0 = 0.f, sh0 = 0.f, sc1 = 0.f, sh1 = 0.f;
    if (MODE) {
      const int c0 = cb + (hi << 1);
      sc0 = bscale[c0];     sh0 = bshift[c0];
      sc1 = bscale[c0 + 1]; sh1 = bshift[c0 + 1];
    }

#pragma unroll
    for (int k = 0; k < KW; ++k) {
      float x0 = sp[k][a0];
      float x1 = sp[k][a1];
      if (MODE) {
        x0 = fmaxf(fmaf(sc0, x0, sh0), 0.f);
        x1 = fmaxf(fmaf(sc1, x1, sh1), 0.f);
      }
      x0 = mk[k] ? x0 : 0.f;
      x1 = mk[k] ? x1 : 0.f;
      v2f bv; bv[0] = x0; bv[1] = x1;

#pragma unroll
      for (int t = 0; t < OT; ++t) {
        v2f av;
        av[0] = wt[t][woff0 + k];
        av[1] = wt[t][woff1 + k];
        acc[t] = wmma4(av, bv, acc[t]);
      }
    }
    a0 += 4 * crow;
    a1 += 4 * crow;
    woff0 += 4 * KW;
    woff1 += 4 * KW;
  }

  // ---- tail block: remaining C_in % 4 channels (runs once) ----------------
  if (C_in & 3) {
    const int  c0  = cbFull + (hi << 1);
    const int  c1  = c0 + 1;
    const bool v0  = (c0 < C_in);
    const bool v1  = (c1 < C_in);
    const int  c0c = v0 ? c0 : (C_in - 1);
    const int  c1c = v1 ? c1 : (C_in - 1);
    const size_t t0 = (size_t)c0c * crow;
    const size_t t1 = (size_t)c1c * crow;
    const int  wo0 = c0c * KW;
    const int  wo1 = c1c * KW;

    float sc0 = 0.f, sh0 = 0.f, sc1 = 0.f, sh1 = 0.f;
    if (MODE) {
      sc0 = bscale[c0c]; sh0 = bshift[c0c];
      sc1 = bscale[c1c]; sh1 = bshift[c1c];
    }

#pragma unroll
    for (int k = 0; k < KW; ++k) {
      float x0 = sp[k][t0];
      float x1 = sp[k][t1];
      if (MODE) {
        x0 = fmaxf(fmaf(sc0, x0, sh0), 0.f);
        x1 = fmaxf(fmaf(sc1, x1, sh1), 0.f);
      }
      x0 = (v0 && mk[k]) ? x0 : 0.f;
      x1 = (v1 && mk[k]) ? x1 : 0.f;
      v2f bv; bv[0] = x0; bv[1] = x1;

#pragma unroll
      for (int t = 0; t < OT; ++t) {
        v2f av;
        av[0] = wt[t][wo0 + k];
        av[1] = wt[t][wo1 + k];
        acc[t] = wmma4(av, bv, acc[t]);
      }
    }
  }

  // D layout: VGPR j -> row M = j + 8*hi (output channel), col N = lo (batch)
#pragma unroll
  for (int t = 0; t < OT; ++t) {
#pragma unroll
    for (int j = 0; j < 8; ++j) {
      const int o = t * 16 + j + (hi << 3);
      if (o < C_out) {
        const size_t ofs = ((size_t)o * N_out + n) * BATCH + b0 + lo;
        float v = acc[t][j] + bias[o];
        if (resid) v += resid[ofs];
        dst[ofs] = v;
      }
    }
  }
}

// ---------------------------------------------------------------------------
// Fused bn4 + relu + mean over nodes:  pooled[c*BATCH+b] = mean_n relu(s*a+t)
// ---------------------------------------------------------------------------
__global__ __launch_bounds__(256)
void k_pool(const float* __restrict__ a, const float* __restrict__ sc,
            const float* __restrict__ sh, float* __restrict__ pooled, int N) {
  const int c = blockIdx.x;
  const int b = threadIdx.x;
  const float s = sc[c], t = sh[c];
  float acc = 0.f;
  for (int n = 0; n < N; ++n)
    acc += fmaxf(fmaf(s, a[((size_t)c * N + n) * BATCH + b], t), 0.f);
  pooled[c * BATCH + b] = acc / (float)N;
}

// ---------------------------------------------------------------------------
// Linear head (69 -> 10) + log_softmax. One block of 256 threads (= batch).
// ---------------------------------------------------------------------------
__global__ __launch_bounds__(256)
void k_head(const float* __restrict__ pooled, const float* __restrict__ w,
            const float* __restrict__ bias, float* __restrict__ out) {
  const int b = threadIdx.x;
  float l[10];
#pragma unroll
  for (int j = 0; j < 10; ++j) {
    float s = bias[j];
    for (int c = 0; c < 69; ++c)
      s = fmaf(pooled[c * BATCH + b], w[j * 69 + c], s);
    l[j] = s;
  }
  float mx = l[0];
#pragma unroll
  for (int j = 1; j < 10; ++j) mx = fmaxf(mx, l[j]);
  float se = 0.f;
#pragma unroll
  for (int j = 0; j < 10; ++j) se += __expf(l[j] - mx);
  const float lse = logf(se);
#pragma unroll
  for (int j = 0; j < 10; ++j) out[b * 10 + j] = l[j] - mx - lse;
}

// ---------------------------------------------------------------------------
// Host-side helpers
// ---------------------------------------------------------------------------
template <int KW, int MODE, int OT>
static inline void launch_conv(const float* src, const int* idx, const float* w,
                               const float* b, const float* sc, const float* sh,
                               const float* resid, float* dst,
                               int Ci, int Ni, int Co, int No, hipStream_t stream) {
  dim3 g(No, 4, 1), blk(32, 4);
  k_ic_conv<KW, MODE, OT><<<g, blk, 0, stream>>>(src, idx, w, b, sc, sh,
                                                 resid, dst, Ci, Ni, Co, No);
}

extern "C" void kernel_launch(void* const* d_in, const int* in_sizes, int n_in,
                              void* d_out, int out_size, void* d_ws, size_t ws_size,
                              hipStream_t stream) {
  (void)in_sizes; (void)n_in; (void)out_size; (void)ws_size;

  const int N0 = 1855, N1 = 1855, N2 = 464, N3 = 116;
  const int n1 = 17, n2 = 35, n3 = 69;

  const float* x       = (const float*)d_in[0];
  const int*   idx0    = (const int*)d_in[1];
  const int*   idx1    = (const int*)d_in[2];
  const int*   idx_r12 = (const int*)d_in[3];
  const int*   idx_12  = (const int*)d_in[4];
  const int*   idx2    = (const int*)d_in[5];
  const int*   idx_r23 = (const int*)d_in[6];
  const int*   idx_23  = (const int*)d_in[7];
  const int*   idx3    = (const int*)d_in[8];

  // Params flattened in JAX sorted-dict order; within sub-dicts: b<w, bt<g.
  int pi = 9;
  auto nxt = [&]() { return (const float*)d_in[pi++]; };
  const float *bn1_bt[8], *bn1_g[8], *bn2_bt[8], *bn2_g[8], *bn3_bt[8], *bn3_g[8];
  for (int i = 0; i < 8; ++i) { bn1_bt[i] = nxt(); bn1_g[i] = nxt(); }
  for (int i = 0; i < 8; ++i) { bn2_bt[i] = nxt(); bn2_g[i] = nxt(); }
  for (int i = 0; i < 8; ++i) { bn3_bt[i] = nxt(); bn3_g[i] = nxt(); }
  const float* bn4_bt  = nxt(); const float* bn4_g  = nxt();
  const float* conv4_b = nxt(); const float* conv4_w = nxt();
  const float* cv0_b   = nxt(); const float* cv0_w   = nxt();
  const float *cv1_b[8], *cv1_w[8], *cv2_b[8], *cv2_w[8], *cv3_b[8], *cv3_w[8];
  for (int i = 0; i < 8; ++i) { cv1_b[i] = nxt(); cv1_w[i] = nxt(); }
  for (int i = 0; i < 8; ++i) { cv2_b[i] = nxt(); cv2_w[i] = nxt(); }
  for (int i = 0; i < 8; ++i) { cv3_b[i] = nxt(); cv3_w[i] = nxt(); }
  const float* res12_b = nxt(); const float* res12_w = nxt();
  const float* res23_b = nxt(); const float* res23_w = nxt();

  // Workspace layout (fp32), ~120 MB total.
  float* wsf = (float*)d_ws;
  size_t off = 0;
  auto alloc = [&](size_t nelem) { float* p = wsf + off; off += nelem; return p; };
  float* xT       = alloc((size_t)3  * N0 * BATCH);
  float* rA       = alloc((size_t)n1 * N1 * BATCH);   // stage-1 residual
  float* hB       = alloc((size_t)n1 * N1 * BATCH);   // stage-1 temp
  float* rC       = alloc((size_t)n2 * N2 * BATCH);   // stage-2 residual
  float* hD       = alloc((size_t)n2 * N2 * BATCH);   // stage-2 temp
  float* rE       = alloc((size_t)n3 * N3 * BATCH);   // stage-3 residual
  float* hF       = alloc((size_t)n3 * N3 * BATCH);   // stage-3 temp
  float* pooled   = alloc((size_t)n3 * BATCH);
  float* scaleBuf = alloc(128);
  float* shiftBuf = alloc(128);

  auto bnstats = [&](const float* a, int C, int NB, const float* g, const float* bt) {
    k_bn_stats<<<C, 256, 0, stream>>>(a, NB, g, bt, scaleBuf, shiftBuf);
  };

  // ---- input transpose + conv0 (C_out=17 -> OT=2) -------------------------
  k_transpose<<<3 * N0, 256, 0, stream>>>(x, xT, 3, N0);
  launch_conv<7, 0, 2>(xT, idx0, cv0_w, cv0_b, nullptr, nullptr, nullptr, rA,
                       3, N0, n1, N1, stream);

  // ---- stage 1: 4 residual pairs (17ch, OT=2) -----------------------------
  for (int i = 0; i < 4; ++i) {
    bnstats(rA, n1, N1 * BATCH, bn1_g[2 * i], bn1_bt[2 * i]);
    launch_conv<7, 1, 2>(rA, idx1, cv1_w[2 * i], cv1_b[2 * i], scaleBuf, shiftBuf,
                         nullptr, hB, n1, N1, n1, N1, stream);
    bnstats(hB, n1, N1 * BATCH, bn1_g[2 * i + 1], bn1_bt[2 * i + 1]);
    launch_conv<7, 1, 2>(hB, idx1, cv1_w[2 * i + 1], cv1_b[2 * i + 1], scaleBuf, shiftBuf,
                         rA, rA, n1, N1, n1, N1, stream);
  }

  // ---- transition 1 -> 2 (C_out=35 -> OT=3) -------------------------------
  launch_conv<1, 0, 3>(rA, idx_r12, res12_w, res12_b, nullptr, nullptr, nullptr, rC,
                       n1, N1, n2, N2, stream);
  bnstats(rA, n1, N1 * BATCH, bn2_g[0], bn2_bt[0]);
  launch_conv<7, 1, 3>(rA, idx_12, cv2_w[0], cv2_b[0], scaleBuf, shiftBuf,
                       nullptr, hD, n1, N1, n2, N2, stream);
  bnstats(hD, n2, N2 * BATCH, bn2_g[1], bn2_bt[1]);
  launch_conv<7, 1, 3>(hD, idx2, cv2_w[1], cv2_b[1], scaleBuf, shiftBuf,
                       rC, rC, n2, N2, n2, N2, stream);

  // ---- stage 2: 3 residual pairs (35ch, OT=3) -----------------------------
  for (int i = 0; i < 3; ++i) {
    bnstats(rC, n2, N2 * BATCH, bn2_g[2 * i + 2], bn2_bt[2 * i + 2]);
    launch_conv<7, 1, 3>(rC, idx2, cv2_w[2 * i + 2], cv2_b[2 * i + 2], scaleBuf, shiftBuf,
                         nullptr, hD, n2, N2, n2, N2, stream);
    bnstats(hD, n2, N2 * BATCH, bn2_g[2 * i + 3], bn2_bt[2 * i + 3]);
    launch_conv<7, 1, 3>(hD, idx2, cv2_w[2 * i + 3], cv2_b[2 * i + 3], scaleBuf, shiftBuf,
                         rC, rC, n2, N2, n2, N2, stream);
  }

  // ---- transition 2 -> 3 (C_out=69 -> OT=5) -------------------------------
  launch_conv<1, 0, 5>(rC, idx_r23, res23_w, res23_b, nullptr, nullptr, nullptr, rE,
                       n2, N2, n3, N3, stream);
  bnstats(rC, n2, N2 * BATCH, bn3_g[0], bn3_bt[0]);
  launch_conv<7, 1, 5>(rC, idx_23, cv3_w[0], cv3_b[0], scaleBuf, shiftBuf,
                       nullptr, hF, n2, N2, n3, N3, stream);
  bnstats(hF, n3, N3 * BATCH, bn3_g[1], bn3_bt[1]);
  launch_conv<7, 1, 5>(hF, idx3, cv3_w[1], cv3_b[1], scaleBuf, shiftBuf,
                       rE, rE, n3, N3, n3, N3, stream);

  // ---- stage 3: 3 residual pairs (69ch, OT=5) -----------------------------
  for (int i = 0; i < 3; ++i) {
    bnstats(rE, n3, N3 * BATCH, bn3_g[2 * i + 2], bn3_bt[2 * i + 2]);
    launch_conv<7, 1, 5>(rE, idx3, cv3_w[2 * i + 2], cv3_b[2 * i + 2], scaleBuf, shiftBuf,
                         nullptr, hF, n3, N3, n3, N3, stream);
    bnstats(hF, n3, N3 * BATCH, bn3_g[2 * i + 3], bn3_bt[2 * i + 3]);
    launch_conv<7, 1, 5>(hF, idx3, cv3_w[2 * i + 3], cv3_b[2 * i + 3], scaleBuf, shiftBuf,
                         rE, rE, n3, N3, n3, N3, stream);
  }

  // ---- head: bn4 + relu + mean-pool + linear + log_softmax ----------------
  bnstats(rE, n3, N3 * BATCH, bn4_g, bn4_bt);
  k_pool<<<n3, 256, 0, stream>>>(rE, scaleBuf, shiftBuf, pooled, N3);
  k_head<<<1, 256, 0, stream>>>(pooled, conv4_w, conv4_b, (float*)d_out);
}